// FullAttnResStack_3977139716411
// MI455X (gfx1250) — compile-verified
//
#include <hip/hip_runtime.h>
#include <math.h>

// Problem constants (match reference)
#define BB   2
#define TT   1024
#define DD   1024
#define LL   12
#define EE   4
#define ND   (EE * DD)     // 4096
#define NTOK (BB * TT)     // 2048
#define EPSF 1e-5f

#define KTILE 32           // k-depth staged per TDM transfer
#define BROW  80           // padded LDS row stride for B (64 data + 16 pad dw)
#define AROW  36           // padded LDS row stride for A (32 data + 4 pad dw)

typedef __attribute__((ext_vector_type(2))) float v2f;
typedef __attribute__((ext_vector_type(8))) float v8f;
typedef __attribute__((ext_vector_type(4))) unsigned int u32x4;
typedef __attribute__((ext_vector_type(4))) int i32x4;
typedef __attribute__((ext_vector_type(8))) int i32x8;
typedef __attribute__((address_space(3))) float lds_fp;

#if defined(__has_builtin)
#  if __has_builtin(__builtin_amdgcn_tensor_load_to_lds) && \
      __has_builtin(__builtin_amdgcn_s_wait_tensorcnt)
#    define HAVE_TDM 1
#  endif
#endif
#ifndef HAVE_TDM
#  define HAVE_TDM 0
#  warning "CDNA5 TDM builtins unavailable: ffn_gemm uses synchronous LDS staging fallback"
#endif

// ---------------------------------------------------------------------------
// Block-wide sum over 256 threads (8 wave32 waves). Returns total to all lanes.
// ---------------------------------------------------------------------------
__device__ __forceinline__ float block_sum(float v, float* sm /*>= 8 floats*/) {
  #pragma unroll
  for (int o = 16; o > 0; o >>= 1) v += __shfl_down(v, o, 32);
  const int wave = threadIdx.x >> 5;
  const int lane = threadIdx.x & 31;
  if (lane == 0) sm[wave] = v;
  __syncthreads();
  float t = (lane < 8) ? sm[lane] : 0.0f;   // each wave reduces redundantly
  #pragma unroll
  for (int o = 4; o > 0; o >>= 1) t += __shfl_down(t, o, 32);
  t = __shfl(t, 0, 32);
  __syncthreads();                           // protect sm for next call
  return t;
}

// ---------------------------------------------------------------------------
// Per-token fused: LN-keys -> scores -> softmax(history) -> residual, u=LN(h)
// One workgroup per token; all history vectors cached in LDS (single HBM read).
// ---------------------------------------------------------------------------
__global__ __launch_bounds__(256)
void attn_kernel(const float* __restrict__ hist, int l,
                 const float* __restrict__ q,
                 const float* __restrict__ kng, const float* __restrict__ knb,
                 const float* __restrict__ lng, const float* __restrict__ lnb,
                 float* __restrict__ resid, float* __restrict__ u) {
  __shared__ float sv[LL * DD];   // 48 KB history cache (l+1 <= 12 vectors)
  __shared__ float sm[8];
  __shared__ float ssc[LL];

  const int t   = blockIdx.x;
  const int tid = threadIdx.x;

  float gq[4], lg[4], lb[4];
  float bq_part = 0.0f, gq_part = 0.0f;
  #pragma unroll
  for (int i = 0; i < 4; ++i) {
    const int d = tid + (i << 8);
    const float qd = q[d];
    gq[i] = kng[d] * qd;
    bq_part += knb[d] * qd;
    gq_part += gq[i];
    lg[i] = lng[d];
    lb[i] = lnb[d];
  }
  const float sum_gq = block_sum(gq_part, sm);
  const float sum_bq = block_sum(bq_part, sm);

  const float scale = 0.03125f;   // 1/sqrt(1024)
  float mu_l = 0.0f, rs_l = 0.0f;

  for (int j = 0; j <= l; ++j) {
    const float* V = hist + (size_t)j * NTOK * DD + (size_t)t * DD;
    float s1 = 0.0f, s2 = 0.0f, sd = 0.0f;
    #pragma unroll
    for (int i = 0; i < 4; ++i) {
      const int d = tid + (i << 8);
      const float v = V[d];
      sv[j * DD + d] = v;
      s1 += v; s2 += v * v; sd += v * gq[i];
    }
    s1 = block_sum(s1, sm);
    s2 = block_sum(s2, sm);
    sd = block_sum(sd, sm);
    const float mu  = s1 * (1.0f / DD);
    const float var = s2 * (1.0f / DD) - mu * mu;
    const float rs  = rsqrtf(var + EPSF);
    if (tid == 0) ssc[j] = ((sd - mu * sum_gq) * rs + sum_bq) * scale;
    if (j == l) { mu_l = mu; rs_l = rs; }
  }
  __syncthreads();

  float mx = -1e30f;
  for (int j = 0; j <= l; ++j) mx = fmaxf(mx, ssc[j]);
  float alpha[LL];
  float den = 0.0f;
  for (int j = 0; j <= l; ++j) { const float e = __expf(ssc[j] - mx); alpha[j] = e; den += e; }
  const float inv = 1.0f / den;

  #pragma unroll
  for (int i = 0; i < 4; ++i) {
    const int d = tid + (i << 8);
    float r = 0.0f;
    for (int j = 0; j <= l; ++j) r += alpha[j] * sv[j * DD + d];
    resid[(size_t)t * DD + d] = r * inv;
    const float h = sv[l * DD + d];
    u[(size_t)t * DD + d] = (h - mu_l) * rs_l * lg[i] + lb[i];
  }
}

// ---------------------------------------------------------------------------
// TDM: DMA a rows x cols f32 tile into LDS with row padding (D# per ISA 08
// sec 8: group0 = {count=1, lds_addr, 57b global_addr, type=2}; group1 =
// {data_size=4B, pad_enable, pad codes, tensor dims = tile dims, stride0}).
// 2-D tensor -> descriptor groups 2/3 zero. Completion via TENSORcnt.
// ---------------------------------------------------------------------------
__device__ __forceinline__ void tdm_load_tile(const float* gsrc, unsigned lds_byte_off,
                                              unsigned cols, unsigned rows,
                                              unsigned row_stride_elems,
                                              unsigned pad_interval_code,
                                              unsigned pad_amount_code) {
#if HAVE_TDM
  const unsigned long long ga = (unsigned long long)(size_t)gsrc;
  u32x4 g0;
  g0[0] = 1u;                                            // count=1, user mode
  g0[1] = lds_byte_off;                                  // lds_addr [63:32]
  g0[2] = (unsigned)ga;                                  // global_addr [95:64]
  g0[3] = (unsigned)((ga >> 32) & 0x01FFFFFFu)           // global_addr [120:96]
        | 0x80000000u;                                   // type=2 [127:126]
  i32x8 g1;
  g1[0] = (int)((2u << 16)                               // data_size = 4 bytes
              | (1u << 20)                               // pad_enable
              | (pad_interval_code << 22)
              | (pad_amount_code << 25));
  g1[1] = (int)(cols << 16);                             // tensor_dim0 lo16
  g1[2] = (int)(rows << 16);                             // dim0 hi=0 | tensor_dim1 lo16
  g1[3] = (int)(cols << 16);                             // dim1 hi=0 | tile_dim0
  g1[4] = (int)rows;                                     // tile_dim1 | tile_dim2=0
  g1[5] = (int)row_stride_elems;                         // tensor_dim0_stride lo32
  g1[6] = 0;                                             // stride0 hi | stride1 lo
  g1[7] = 0;                                             // stride1 hi (unused, 2-D)
  i32x4 z4 = {0, 0, 0, 0};
#if __clang_major__ >= 23
  i32x8 z8 = {0, 0, 0, 0, 0, 0, 0, 0};
  __builtin_amdgcn_tensor_load_to_lds(g0, g1, z4, z4, z8, 0);
#else
  __builtin_amdgcn_tensor_load_to_lds(g0, g1, z4, z4, 0);
#endif
#else
  (void)gsrc; (void)lds_byte_off; (void)cols; (void)rows;
  (void)row_stride_elems; (void)pad_interval_code; (void)pad_amount_code;
#endif
}

// ---------------------------------------------------------------------------
// GEMM C = op(A @ Bm + bias [+ resid]) via V_WMMA_F32_16X16X4_F32.
// Wave tile 16(M)x64(N); block = 8 waves -> 128x64. BOTH A (128xKTILE) and B
// (KTILEx64) tiles are DMA'd into LDS by the Tensor Data Mover, double
// buffered; the inner loop is pure LDS + WMMA (no VMEM). Padded LDS strides
// (AROW=36, BROW=80) make every ds read bank-conflict-free.
// Grids divide exactly -> EXEC all-ones for every WMMA.
// ---------------------------------------------------------------------------
__global__ __launch_bounds__(256)
void ffn_gemm(const float* __restrict__ A, const float* __restrict__ Bm,
              const float* __restrict__ bias, const float* __restrict__ resid,
              float* __restrict__ C, int M, int N, int K, int fuse_gelu) {
  __shared__ float Asm[2][128 * AROW];      // 2 x 18 KB
  __shared__ float Bsm[2][KTILE * BROW];    // 2 x 10 KB

  const int tid  = threadIdx.x;
  const int lane = tid & 31;
  const int wave = tid >> 5;
  const int mblk = blockIdx.y * 128;
  const int m0 = mblk + wave * 16;
  const int n0 = blockIdx.x * 64;
  const int r  = lane & 15;
  const int kk = (lane >> 4) << 1;          // 0 or 2
  (void)M;

  v8f acc[4];
  #pragma unroll
  for (int nt = 0; nt < 4; ++nt)
    #pragma unroll
    for (int e = 0; e < 8; ++e) acc[nt][e] = 0.0f;

#if HAVE_TDM
  const unsigned ldsA0 = (unsigned)(size_t)(lds_fp*)&Asm[0][0];
  const unsigned ldsB0 = (unsigned)(size_t)(lds_fp*)&Bsm[0][0];
  const unsigned strideA = (unsigned)(128 * AROW * sizeof(float));
  const unsigned strideB = (unsigned)(KTILE * BROW * sizeof(float));
  if (wave == 0) {   // prime stage 0 (A then B; in-order per wave)
    tdm_load_tile(A + (size_t)mblk * K, ldsA0, KTILE, 128, K, /*32dw*/4, /*+4dw*/3);
    tdm_load_tile(Bm + n0,              ldsB0, 64, KTILE, N, /*64dw*/5, /*+16dw*/15);
  }
#endif

  const int nstage = K / KTILE;
  for (int kt = 0; kt < nstage; ++kt) {
    const int k0 = kt * KTILE;
#if HAVE_TDM
    const int buf = kt & 1;
    if (wave == 0) {
      if (kt + 1 < nstage) {
        const unsigned nb = (unsigned)((kt + 1) & 1);
        tdm_load_tile(A + (size_t)mblk * K + (k0 + KTILE),
                      ldsA0 + nb * strideA, KTILE, 128, K, 4, 3);
        tdm_load_tile(Bm + (size_t)(k0 + KTILE) * N + n0,
                      ldsB0 + nb * strideB, 64, KTILE, N, 5, 15);
        __builtin_amdgcn_s_wait_tensorcnt(2);  // stage kt's pair complete
      } else {
        __builtin_amdgcn_s_wait_tensorcnt(0);
      }
    }
    __syncthreads();                           // LDS tiles visible to all waves
#else
    const int buf = 0;
    __syncthreads();                           // prior reads of tiles done
    for (int idx = tid; idx < 128 * (KTILE / 4); idx += 256) {
      const int row = idx / (KTILE / 4);
      const int c4  = (idx % (KTILE / 4)) * 4;
      *reinterpret_cast<float4*>(&Asm[0][row * AROW + c4]) =
          *reinterpret_cast<const float4*>(A + (size_t)(mblk + row) * K + k0 + c4);
    }
    for (int idx = tid; idx < KTILE * 16; idx += 256) {
      const int row = idx >> 4;
      const int c4  = (idx & 15) << 2;
      *reinterpret_cast<float4*>(&Bsm[0][row * BROW + c4]) =
          *reinterpret_cast<const float4*>(Bm + (size_t)(k0 + row) * N + n0 + c4);
    }
    __syncthreads();
#endif

    const float* as = &Asm[buf][(wave * 16 + r) * AROW + kk];
    const float* bs = &Bsm[buf][0];
    #pragma unroll
    for (int k4 = 0; k4 < KTILE; k4 += 4) {
      v2f a;
      a.x = as[k4];
      a.y = as[k4 + 1];
      const float* brow0 = bs + (k4 + kk) * BROW + r;
      const float* brow1 = brow0 + BROW;
      #pragma unroll
      for (int nt = 0; nt < 4; ++nt) {
        v2f b;
        b.x = brow0[nt * 16];
        b.y = brow1[nt * 16];
        acc[nt] = __builtin_amdgcn_wmma_f32_16x16x4_f32(
            false, a, false, b, (short)0, acc[nt], false, false);
      }
    }
    __syncthreads();   // all waves done with buf before TDM refills it
  }

  const int rbase = m0 + ((lane >> 4) << 3);
  #pragma unroll
  for (int nt = 0; nt < 4; ++nt) {
    const int col = n0 + nt * 16 + r;
    const float bv = bias[col];
    #pragma unroll
    for (int i = 0; i < 8; ++i) {
      const int row = rbase + i;
      float val = acc[nt][i] + bv;
      if (fuse_gelu) {
        val = 0.5f * val * (1.0f + erff(val * 0.70710678118654752f));  // exact GELU
      } else {
        val += resid[(size_t)row * N + col];
      }
      C[(size_t)row * N + col] = val;
    }
  }
}

// ---------------------------------------------------------------------------
// Host-side orchestration: 12 layers x (attn, GEMM1+gelu, GEMM2+residual).
// ws layout: hist[12 x 8MB] | u (8MB) | resid (8MB) | act (32MB) = 144MB
// ---------------------------------------------------------------------------
extern "C" void kernel_launch(void* const* d_in, const int* in_sizes, int n_in,
                              void* d_out, int out_size, void* d_ws, size_t ws_size,
                              hipStream_t stream) {
  (void)in_sizes; (void)n_in; (void)out_size; (void)ws_size;
  const float* x   = (const float*)d_in[0];
  const float* q   = (const float*)d_in[1];
  const float* kng = (const float*)d_in[2];
  const float* knb = (const float*)d_in[3];
  const float* lng = (const float*)d_in[4];
  const float* lnb = (const float*)d_in[5];
  const float* W1  = (const float*)d_in[6];
  const float* b1  = (const float*)d_in[7];
  const float* W2  = (const float*)d_in[8];
  const float* b2  = (const float*)d_in[9];

  float* ws = (float*)d_ws;
  const size_t HSZ = (size_t)NTOK * DD;
  float* hist  = ws;                     // 12 history slots
  float* u     = hist + 12 * HSZ;
  float* resid = u + HSZ;
  float* act   = resid + HSZ;            // NTOK * ND

  hipMemcpyAsync(hist, x, HSZ * sizeof(float), hipMemcpyDeviceToDevice, stream);

  for (int l = 0; l < LL; ++l) {
    attn_kernel<<<NTOK, 256, 0, stream>>>(hist, l,
        q + (size_t)l * DD, kng + (size_t)l * DD, knb + (size_t)l * DD,
        lng + (size_t)l * DD, lnb + (size_t)l * DD, resid, u);

    ffn_gemm<<<dim3(ND / 64, NTOK / 128), 256, 0, stream>>>(
        u, W1 + (size_t)l * DD * ND, b1 + (size_t)l * ND, nullptr,
        act, NTOK, ND, DD, /*fuse_gelu=*/1);

    float* hout = (l == LL - 1) ? (float*)d_out : hist + (size_t)(l + 1) * HSZ;
    ffn_gemm<<<dim3(DD / 64, NTOK / 128), 256, 0, stream>>>(
        act, W2 + (size_t)l * ND * DD, b2 + (size_t)l * DD, resid,
        hout, NTOK, DD, ND, /*fuse_gelu=*/0);
  }
}